// FBSNN_70884140253988
// MI455X (gfx1250) — compile-verified
//
#include <hip/hip_runtime.h>

// ---------------------------------------------------------------------------
// FBSNN (LQ control) loss on gfx1250.
// Per wave: 16 SDE paths; MLP layers done with V_WMMA_F32_16X16X4_F32.
// Weights + per-wave activation staging in LDS (row stride 68 dwords ->
// conflict-free b64 loads, biases folded in as K=64 column).
// ---------------------------------------------------------------------------

typedef float v2f __attribute__((ext_vector_type(2)));
typedef float v8f __attribute__((ext_vector_type(8)));

#define STEPS    100
#define DT       0.01f
#define SQRT_DT  0.1f
#define SIGMA    0.5f
#define ALPHA    1.0f

#define WAVE          32
#define WAVES_PER_WG  8
#define THREADS       (WAVE * WAVES_PER_WG)
#define SPW           16                     // samples per wave (WMMA N tile)
#define WG_SAMPLES    (WAVES_PER_WG * SPW)

#define WROW  68                             // LDS row stride in dwords (64 W + bias + 3 zero)
#define WMAT  (64 * WROW)                    // one 64x64(+bias) weight block
#define STAGE_PER_WAVE (2 * 16 * WROW)       // h and dh stage, 16 samples x 68 cols

// LDS layout (float offsets)
#define OFF_W2    0
#define OFF_W3    (OFF_W2 + WMAT)
#define OFF_W4    (OFF_W3 + WMAT)
#define OFF_W1    (OFF_W4 + WMAT)            // layer-1: 64 rows x 4 (w_t, w_y, b, 0)
#define OFF_W5    (OFF_W1 + 256)             // output weights (64)
#define OFF_PART  (OFF_W5 + 64)              // per-wave partial losses
#define OFF_STAGE (OFF_PART + WAVES_PER_WG)
#define SMEM_FLOATS (OFF_STAGE + WAVES_PER_WG * STAGE_PER_WAVE)

template <int MASK>
__device__ __forceinline__ float swz(float v) {
  // ds_swizzle group-of-32 mode: and=0x1f, or=0, xor=MASK
  return __int_as_float(
      __builtin_amdgcn_ds_swizzle(__float_as_int(v), (MASK << 10) | 0x1F));
}

__device__ __forceinline__ v8f wmma4(v2f a, v2f b, v8f c) {
  // D(16x16,f32) = A(16x4,f32) x B(4x16,f32) + C
  return __builtin_amdgcn_wmma_f32_16x16x4_f32(
      false, a, false, b, (short)0, c, false, false);
}

// One 64x64(+bias) hidden layer: 17 k-tiles (K=4) x 4 M-tiles, fwd + tangent.
__device__ __forceinline__ void layer_hidden(const float* __restrict__ wb,
                                             const float* __restrict__ sH,
                                             const float* __restrict__ sD,
                                             int m, int khalf,
                                             v8f acc[4], v8f accd[4]) {
  const v8f z = {0.f, 0.f, 0.f, 0.f, 0.f, 0.f, 0.f, 0.f};
#pragma unroll
  for (int T = 0; T < 4; ++T) { acc[T] = z; accd[T] = z; }
#pragma unroll
  for (int kt = 0; kt < 17; ++kt) {
    const int col = 4 * kt + khalf;                // even -> 8B aligned
    v2f bh = *(const v2f*)(sH + m * WROW + col);   // B layout: K in lane halves
    v2f bd = *(const v2f*)(sD + m * WROW + col);
#pragma unroll
    for (int T = 0; T < 4; ++T) {
      v2f a = *(const v2f*)(wb + (16 * T + m) * WROW + col);  // A layout
      acc[T]  = wmma4(a, bh, acc[T]);
      accd[T] = wmma4(a, bd, accd[T]);
    }
  }
}

// sin activation + JVP, then store D-layout tiles back to the wave stage.
__device__ __forceinline__ void act_store(v8f acc[4], v8f accd[4],
                                          float* __restrict__ sH,
                                          float* __restrict__ sD,
                                          int m, int nb) {
#pragma unroll
  for (int T = 0; T < 4; ++T) {
    float hv[8], dv[8];
#pragma unroll
    for (int r = 0; r < 8; ++r) {
      float pre = acc[T][r];
      hv[r] = __sinf(pre);
      dv[r] = __cosf(pre) * accd[T][r];
    }
    float* ph = sH + m * WROW + 16 * T + nb;  // 16B aligned (272m + 64T + {0,32})
    float* pd = sD + m * WROW + 16 * T + nb;
    *(float4*)ph       = make_float4(hv[0], hv[1], hv[2], hv[3]);
    *(float4*)(ph + 4) = make_float4(hv[4], hv[5], hv[6], hv[7]);
    *(float4*)pd       = make_float4(dv[0], dv[1], dv[2], dv[3]);
    *(float4*)(pd + 4) = make_float4(dv[4], dv[5], dv[6], dv[7]);
  }
}

// Full net + dY/dy for this wave's 16 samples (scalar per lane, halves dup).
__device__ __forceinline__ void eval_net(
    float t, float y, const float* w1b, const float* w2b, const float* w3b,
    const float* w4b, float* sH, float* sD, const float (&w5s)[32], float b5,
    int lane, float& Y, float& dY) {
  const int  m     = lane & 15;
  const bool hic   = (lane >= 16);
  const int  khalf = hic ? 2 : 0;
  const int  nb    = hic ? 8 : 0;
  const v8f  z     = {0.f, 0.f, 0.f, 0.f, 0.f, 0.f, 0.f, 0.f};
  v8f acc[4], accd[4];

  // Layer 1 via one K=4 WMMA: x = [t, y, 1, 0]; A rows = [w_t, w_y, b, 0].
  v2f bF, bT;
  bF.x = hic ? 1.0f : t;  bF.y = hic ? 0.0f : y;   // lanes>=16 feed K=2,3
  bT.x = 0.0f;            bT.y = hic ? 0.0f : 1.0f; // d/dy tangent = [0,1,0,0]
#pragma unroll
  for (int T = 0; T < 4; ++T) {
    v2f a = *(const v2f*)(w1b + (16 * T + m) * 4 + khalf);
    acc[T]  = wmma4(a, bF, z);
    accd[T] = wmma4(a, bT, z);
  }
  act_store(acc, accd, sH, sD, m, nb);

  layer_hidden(w2b, sH, sD, m, khalf, acc, accd);
  act_store(acc, accd, sH, sD, m, nb);
  layer_hidden(w3b, sH, sD, m, khalf, acc, accd);
  act_store(acc, accd, sH, sD, m, nb);
  layer_hidden(w4b, sH, sD, m, khalf, acc, accd);

  // Output layer (1x64) straight from D-layout registers.
  float p = 0.f, pd = 0.f;
#pragma unroll
  for (int T = 0; T < 4; ++T)
#pragma unroll
    for (int r = 0; r < 8; ++r) {
      float pre = acc[T][r];
      float h = __sinf(pre);
      float d = __cosf(pre) * accd[T][r];
      p  = fmaf(w5s[T * 8 + r], h, p);
      pd = fmaf(w5s[T * 8 + r], d, pd);
    }
  // each sample's 64 partials live in lanes n and n+16 -> xor-16 combine
  Y  = p  + swz<16>(p) + b5;
  dY = pd + swz<16>(pd);
}

__global__ __launch_bounds__(THREADS) void fbsnn_kernel(
    const float* __restrict__ W0, const float* __restrict__ b0,
    const float* __restrict__ W1, const float* __restrict__ b1,
    const float* __restrict__ W2, const float* __restrict__ b2,
    const float* __restrict__ W3, const float* __restrict__ b3,
    const float* __restrict__ W4, const float* __restrict__ b4,
    const float* __restrict__ y0p, const float* __restrict__ dW,
    float* __restrict__ partial, int B) {
  extern __shared__ float smem[];
  const int tid = threadIdx.x;

  // ---- stage weights into LDS (bias folded in as column 64) ----
  {
    const float* Wg[3] = {W1, W2, W3};
    const float* bg[3] = {b1, b2, b3};
    for (int L = 0; L < 3; ++L) {
      float* dst = smem + L * WMAT;
      for (int idx = tid; idx < WMAT; idx += THREADS) {
        int r = idx / WROW, c = idx - r * WROW;
        dst[idx] = (c < 64) ? Wg[L][r * 64 + c]
                            : ((c == 64) ? bg[L][r] : 0.0f);
      }
    }
    if (tid < 256) {  // layer-1 block: [w_t, w_y, b, 0] per row
      int r = tid >> 2, c = tid & 3;
      smem[OFF_W1 + tid] =
          (c == 0) ? W0[r * 2] : ((c == 1) ? W0[r * 2 + 1]
                                           : ((c == 2) ? b0[r] : 0.0f));
    }
    if (tid < 64) smem[OFF_W5 + tid] = W4[tid];
  }
  __syncthreads();

  const int  lane = tid & 31;
  const int  wave = tid >> 5;
  const int  m    = lane & 15;
  const bool hic  = (lane >= 16);

  float* sH = smem + OFF_STAGE + wave * STAGE_PER_WAVE;
  float* sD = sH + 16 * WROW;
  const float* w1b = smem + OFF_W1;
  const float* w2b = smem + OFF_W2;
  const float* w3b = smem + OFF_W3;
  const float* w4b = smem + OFF_W4;

  // constant bias-input columns of the stage (per wave, once)
  if (lane < 16) {
    *(float4*)(sH + lane * WROW + 64) = make_float4(1.f, 0.f, 0.f, 0.f);
    *(float4*)(sD + lane * WROW + 64) = make_float4(0.f, 0.f, 0.f, 0.f);
  }

  // preload output weights into registers in this lane's D-slot order
  float w5s[32];
  {
    const int nb = hic ? 8 : 0;
#pragma unroll
    for (int T = 0; T < 4; ++T)
#pragma unroll
      for (int r = 0; r < 8; ++r)
        w5s[T * 8 + r] = smem[OFF_W5 + 16 * T + r + nb];
  }
  const float b5 = b4[0];

  const int sample = blockIdx.x * WG_SAMPLES + wave * SPW + m;
  float t = 0.f, y = y0p[0];
  float Y, dY;
  eval_net(t, y, w1b, w2b, w3b, w4b, sH, sD, w5s, b5, lane, Y, dY);

  float loss = 0.f;
  for (int s = 0; s < STEPS; ++s) {
    float Z = SIGMA * dY;                          // sigma * dY/dy
    float q = (-0.5f / (SIGMA * SIGMA)) * Z;       // optimal control
    float dws = dW[(size_t)s * B + sample] * SQRT_DT;
    if (s + 1 < STEPS)                              // global_prefetch_b8
      __builtin_prefetch(dW + (size_t)(s + 1) * B + sample, 0, 0);
    float y1 = y + q * DT + SIGMA * dws;           // Euler-Maruyama
    float t1 = t + DT;
    float Y1, dY1;
    eval_net(t1, y1, w1b, w2b, w3b, w4b, sH, sD, w5s, b5, lane, Y1, dY1);
    float Yt = Y - (q * q) * DT + Z * dws;
    float r  = Y1 - Yt;
    loss = fmaf(r, r, loss);
    Y = Y1; dY = dY1; y = y1; t = t1;
  }
  {
    float r1 = Y - ALPHA * y * y;                  // terminal cost residual
    float r2 = dY - 2.0f * ALPHA * y;              // terminal grad residual
    loss = fmaf(r1, r1, loss);
    loss = fmaf(r2, r2, loss);
  }

  // lanes n and n+16 hold identical per-sample loss: reduce 32, halve.
  loss += swz<16>(loss);
  loss += swz<8>(loss);
  loss += swz<4>(loss);
  loss += swz<2>(loss);
  loss += swz<1>(loss);
  loss *= 0.5f;

  if (lane == 0) smem[OFF_PART + wave] = loss;
  __syncthreads();
  if (tid == 0) {
    float sum = 0.f;
    for (int w = 0; w < WAVES_PER_WG; ++w) sum += smem[OFF_PART + w];
    partial[blockIdx.x] = sum;
  }
}

__global__ __launch_bounds__(256) void fbsnn_reduce(
    const float* __restrict__ partial, int np, float invB,
    float* __restrict__ out) {
  __shared__ float red[256];
  int tid = threadIdx.x;
  float s = 0.f;
  int per = np / 256;
  for (int i = 0; i < per; ++i) s += partial[tid * per + i];
  int rem = np - per * 256;
  if (tid < rem) s += partial[per * 256 + tid];
  red[tid] = s;
  __syncthreads();
  for (int off = 128; off > 0; off >>= 1) {
    if (tid < off) red[tid] += red[tid + off];
    __syncthreads();
  }
  if (tid == 0) out[0] = red[0] * invB;
}

extern "C" void kernel_launch(void* const* d_in, const int* in_sizes, int n_in,
                              void* d_out, int out_size, void* d_ws,
                              size_t ws_size, hipStream_t stream) {
  (void)n_in; (void)out_size; (void)ws_size;
  const float* W0 = (const float*)d_in[0];
  const float* b0 = (const float*)d_in[1];
  const float* W1 = (const float*)d_in[2];
  const float* b1 = (const float*)d_in[3];
  const float* W2 = (const float*)d_in[4];
  const float* b2 = (const float*)d_in[5];
  const float* W3 = (const float*)d_in[6];
  const float* b3 = (const float*)d_in[7];
  const float* W4 = (const float*)d_in[8];
  const float* b4 = (const float*)d_in[9];
  const float* y0 = (const float*)d_in[10];
  const float* dW = (const float*)d_in[11];

  const int B      = in_sizes[11] / STEPS;   // 131072
  const int blocks = B / WG_SAMPLES;         // 1024
  float* partial   = (float*)d_ws;
  float* out       = (float*)d_out;

  const size_t shmem = (size_t)SMEM_FLOATS * sizeof(float);  // ~120 KB < 320 KB/WGP
  fbsnn_kernel<<<blocks, THREADS, shmem, stream>>>(
      W0, b0, W1, b1, W2, b2, W3, b3, W4, b4, y0, dW, partial, B);
  fbsnn_reduce<<<1, 256, 0, stream>>>(partial, blocks, 1.0f / (float)B, out);
}